// min_model_85830626443600
// MI455X (gfx1250) — compile-verified
//
#include <hip/hip_runtime.h>

// MI455X (gfx1250) implementation.
//
// Reference collapses (one-hot algebra):
//   q = idx[b,t]
//   attention[b,t,s] = w0*m[s] + w1*m[s-1] + w2*C(s-2),  m[s]=(idx[b,s]==q), C = prefix count of m
//   logits[b,t,v]    = sum_{s<=t, idx[b,s]==v} attention[b,t,s]
//                    = w0*a0[v] + w1*a1[v] + w2*a2[v]   (a* = integer scatter sums)
//   out = softmax_v(logits)
//
// Memory-bound by the 64MB fp32 output (~2.9us at 23.3 TB/s). Scatter uses exact,
// deterministic integer LDS atomics. Softmax denominator uses chained
// V_WMMA_F32_16X16X4_F32 with an all-ones B (exact f32 row-sum trick).

#define TT   2048
#define VV   1024
#define BLK  256
#define NWAVE (BLK / 32)

typedef float v2f __attribute__((ext_vector_type(2)));
typedef float v8f __attribute__((ext_vector_type(8)));

__global__ __launch_bounds__(BLK) void relpos_softmax_kernel(
    const int* __restrict__ idx, const float* __restrict__ w, float* __restrict__ out) {
  __shared__ int   accA[VV];          // low16: a0 (w0 hits), high16: a1 (w1 hits)
  __shared__ int   accB[VV];          // a2 = sum of C(s-2)
  __shared__ float lf[VV];            // exp(logit - max) values
  __shared__ int   wvcnt[NWAVE];
  __shared__ float wred[NWAVE];
  __shared__ float sTotal;

  const int tid  = threadIdx.x;
  const int lane = tid & 31;
  const int wid  = tid >> 5;

  const int bt = blockIdx.x;
  const int b  = bt >> 11;            // / TT
  const int t  = bt & (TT - 1);
  const int* __restrict__ row = idx + b * TT;
  const int q = row[t];

  const float w0 = w[0], w1 = w[1], w2 = w[2];

  for (int v = tid; v < VV; v += BLK) { accA[v] = 0; accB[v] = 0; }
  __syncthreads();

  // ---- scan s in [0, t]: ballot-based prefix count + integer scatter ----
  int carry = 0;
  for (int base = 0; base <= t; base += BLK) {
    const int  s     = base + tid;
    const bool valid = (s <= t);
    int tok = -1, tokp = -1;
    if (valid) {
      tok  = row[s];
      tokp = (s > 0) ? row[s - 1] : -1;
    }
    const bool m  = valid && (tok == q);
    const bool mp = valid && (tokp == q);

    const unsigned long long bal = __ballot(m);        // wave32: low 32 bits
    const int inclw = __popcll(bal & ((2ull << lane) - 1ull));
    if (lane == 0) wvcnt[wid] = __popcll(bal);
    __syncthreads();

    int pre = 0, chunk = 0;
    #pragma unroll
    for (int wv = 0; wv < NWAVE; ++wv) {
      const int c = wvcnt[wv];
      if (wv < wid) pre += c;
      chunk += c;
    }

    if (valid) {
      const int incl = carry + pre + inclw;            // C(s), inclusive
      const int mi = m ? 1 : 0, mpi = mp ? 1 : 0;
      const int a01 = mi | (mpi << 16);
      if (a01) atomicAdd(&accA[tok], a01);             // ds_add_u32: exact, deterministic
      const int c2 = incl - mi - mpi;                  // C(s-2) >= 0
      if (c2) atomicAdd(&accB[tok], c2);
    }
    carry += chunk;
    __syncthreads();                                   // protect wvcnt reuse
  }

  // ---- logits (exact) + block max ----
  float lv[VV / BLK];
  float lmax = -3.0e38f;
  #pragma unroll
  for (int k = 0; k < VV / BLK; ++k) {
    const int v   = tid + k * BLK;
    const int a01 = accA[v];
    const float l = w0 * (float)(a01 & 0xFFFF) + w1 * (float)(a01 >> 16) + w2 * (float)accB[v];
    lv[k] = l;
    lmax  = fmaxf(lmax, l);
  }
  #pragma unroll
  for (int off = 16; off > 0; off >>= 1)
    lmax = fmaxf(lmax, __shfl_xor(lmax, off, 32));
  if (lane == 0) wred[wid] = lmax;
  __syncthreads();
  float gmax = wred[0];
  #pragma unroll
  for (int wv = 1; wv < NWAVE; ++wv) gmax = fmaxf(gmax, wred[wv]);

  // ---- exp ----
  #pragma unroll
  for (int k = 0; k < VV / BLK; ++k) {
    const int v = tid + k * BLK;
    lf[v] = __expf(lv[k] - gmax);
  }
  __syncthreads();

  // ---- denominator: wave0 sums 1024 values via 16 chained f32 WMMAs ----
  // B = all-ones(4x16): D[i][j] = rowsum_i(A) + C[i][j]; grand total is
  // layout-invariant, recovered from one D column (lanes 0 and 16).
  if (tid < 32) {
    const int r  = lane & 15;
    const int kb = (lane < 16) ? 0 : 2;
    v8f acc = {};
    v2f bones; bones[0] = 1.0f; bones[1] = 1.0f;
    #pragma unroll
    for (int c = 0; c < 16; ++c) {
      v2f a;
      a[0] = lf[c * 64 + r * 4 + kb];
      a[1] = lf[c * 64 + r * 4 + kb + 1];
      acc = __builtin_amdgcn_wmma_f32_16x16x4_f32(
          /*neg_a=*/false, a, /*neg_b=*/false, bones,
          /*c_mod=*/(short)0, acc, /*reuse_a=*/false, /*reuse_b=*/false);
    }
    float p = 0.0f;
    #pragma unroll
    for (int i = 0; i < 8; ++i) p += acc[i];
    const float total = __shfl(p, 0, 32) + __shfl(p, 16, 32);
    if (lane == 0) sTotal = total;
  }
  __syncthreads();

  // ---- normalize + write (64MB stream: the roofline term) ----
  const float inv = 1.0f / sTotal;
  float* __restrict__ orow = out + (size_t)bt * VV;
  #pragma unroll
  for (int k = 0; k < VV / BLK; ++k) {
    const int v = tid + k * BLK;
    orow[v] = lf[v] * inv;
  }
}

extern "C" void kernel_launch(void* const* d_in, const int* in_sizes, int n_in,
                              void* d_out, int out_size, void* d_ws, size_t ws_size,
                              hipStream_t stream) {
  (void)in_sizes; (void)n_in; (void)d_ws; (void)ws_size;
  const int*   idx = (const int*)d_in[0];     // [B, T] int32
  const float* w   = (const float*)d_in[1];   // [4, 1] float32
  float*       out = (float*)d_out;           // [B, T, V] float32
  const int nrows = out_size / VV;            // B*T = 16384
  relpos_softmax_kernel<<<nrows, BLK, 0, stream>>>(idx, w, out);
}